// GCN_66254165508930
// MI455X (gfx1250) — compile-verified
//
#include <hip/hip_runtime.h>
#include <hip/hip_bf16.h>

typedef float v2f __attribute__((ext_vector_type(2)));
typedef float v8f __attribute__((ext_vector_type(8)));

#define TPB 256

// ---------------- utility kernels ----------------

__global__ void k_zero(float* __restrict__ p, int n) {
    int i = blockIdx.x * blockDim.x + threadIdx.x;
    if (i < n) p[i] = 0.0f;
}

// deg[dst] += 1 for each edge (deg buffer pre-zeroed)
__global__ void k_deg(const int* __restrict__ dst, float* __restrict__ deg, int E) {
    int i = blockIdx.x * blockDim.x + threadIdx.x;
    if (i < E) atomicAdd(&deg[dst[i]], 1.0f);
}

// in-place: d[n] = 1/sqrt(deg[n] + 1)
__global__ void k_dinv(float* __restrict__ d, int n) {
    int i = blockIdx.x * blockDim.x + threadIdx.x;
    if (i < n) d[i] = rsqrtf(d[i] + 1.0f);
}

// h[n, :] = embed[x[n], :]   (one wave per node, float4 per lane)
__global__ void k_gather(const int* __restrict__ x, const float* __restrict__ embed,
                         float* __restrict__ h, int n) {
    int t = blockIdx.x * blockDim.x + threadIdx.x;
    int node = t >> 5, q = t & 31;
    if (node >= n) return;
    float4 v = ((const float4*)(embed + (size_t)x[node] * 128))[q];
    ((float4*)(h + (size_t)node * 128))[q] = v;
}

// ---------------- fp32 WMMA GEMM: hw = act(h) @ W  (128x128 W) ----------------
// Each wave: 16 rows x 128 cols. 8 waves/block -> 128 rows per block.
// W is staged in LDS pre-swizzled by K-pair so each lane's B fragment
// (W[k][c], W[k+1][c]) is one contiguous ds_load_b64. ReLU of the previous
// layer is a template parameter (no per-step uniform branches).
template <int RELU>
__global__ __launch_bounds__(TPB)
void k_gemm128(const float* __restrict__ h, const float* __restrict__ W,
               float* __restrict__ hw, int nrows) {
    __shared__ float Ws[128 * 128];  // 64 KB of the 320 KB WGP LDS
    for (int i = threadIdx.x; i < 128 * 128; i += TPB) {
        int k = i >> 7, c = i & 127;
        // pair-swizzled: Ws[(pair*128 + c)*2 + (k&1)] = W[k][c]
        Ws[(((k >> 1) * 128 + c) << 1) + (k & 1)] = W[i];
    }
    __syncthreads();

    const int wave = threadIdx.x >> 5;
    const int lane = threadIdx.x & 31;
    const int half = lane >> 4;   // 0: lanes 0-15, 1: lanes 16-31
    const int l16  = lane & 15;
    const int mt   = blockIdx.x * 8 + wave;
    const int m0   = mt * 16;
    if (m0 >= nrows) return;      // wave-uniform exit: EXEC stays all-1s for WMMA

    int arow = m0 + l16;
    if (arow >= nrows) arow = nrows - 1;  // clamp tail loads; stores predicated below
    // A fragment per ISA 7.12.2: lane<16 holds K=k0,k0+1 ; lane>=16 holds K=k0+2,k0+3
    const float* __restrict__ aptr = h + (size_t)arow * 128 + half * 2;
    __builtin_prefetch(aptr, 0, 1);   // global_prefetch_b8 on the A stream

    v8f acc[8];
#pragma unroll
    for (int t = 0; t < 8; ++t) acc[t] = (v8f){0.f, 0.f, 0.f, 0.f, 0.f, 0.f, 0.f, 0.f};

#pragma unroll
    for (int kk = 0; kk < 128; kk += 32) {
        // batch 8 K-steps of A loads so global latency overlaps WMMA issue
        v2f a[8];
#pragma unroll
        for (int p = 0; p < 8; ++p)
            a[p] = *(const v2f*)(aptr + kk + 4 * p);
#pragma unroll
        for (int p = 0; p < 8; ++p) {
            if (RELU) { a[p].x = fmaxf(a[p].x, 0.f); a[p].y = fmaxf(a[p].y, 0.f); }
            const int pair = ((kk + 4 * p) >> 1) + half;  // K-pair index for this half-wave
            const float* __restrict__ brow = Ws + ((size_t)pair * 128 << 1);
#pragma unroll
            for (int t = 0; t < 8; ++t) {
                v2f b = *(const v2f*)(brow + ((t * 16 + l16) << 1));
                acc[t] = __builtin_amdgcn_wmma_f32_16x16x4_f32(
                    false, a[p], false, b, (short)0, acc[t], false, false);
            }
        }
    }

    // C/D layout: VGPR i -> row m0 + i + 8*half, col = tile*16 + l16
#pragma unroll
    for (int t = 0; t < 8; ++t) {
#pragma unroll
        for (int i = 0; i < 8; ++i) {
            int row = m0 + i + 8 * half;
            if (row < nrows)
                hw[(size_t)row * 128 + t * 16 + l16] = acc[t][i];
        }
    }
}

// agg[n,:] = hw[n,:] * dinv[n]^2 + b[:]    (self-loop term + bias, before scatter)
__global__ void k_agg_init(const float* __restrict__ hw, const float* __restrict__ dinv,
                           const float* __restrict__ bias, float* __restrict__ agg, int n) {
    int t = blockIdx.x * blockDim.x + threadIdx.x;
    int node = t >> 5, q = t & 31;
    if (node >= n) return;
    float di = dinv[node], s = di * di;
    float4 v = ((const float4*)(hw + (size_t)node * 128))[q];
    float4 b = ((const float4*)bias)[q];
    float4 o;
    o.x = v.x * s + b.x; o.y = v.y * s + b.y;
    o.z = v.z * s + b.z; o.w = v.w * s + b.w;
    ((float4*)(agg + (size_t)node * 128))[q] = o;
}

// one wave32 per edge: agg[dst,:] += hw[src,:] * dinv[src]*dinv[dst]
__global__ void k_scatter(const int* __restrict__ src, const int* __restrict__ dst,
                          const float* __restrict__ dinv, const float* __restrict__ hw,
                          float* __restrict__ agg, int E) {
    int t = blockIdx.x * blockDim.x + threadIdx.x;
    int e = t >> 5, q = t & 31;
    if (e >= E) return;
    int s = src[e], d = dst[e];
    float w = dinv[s] * dinv[d];
    float4 v = ((const float4*)(hw + (size_t)s * 128))[q];
    float* ap = agg + (size_t)d * 128 + (size_t)q * 4;
    atomicAdd(ap + 0, v.x * w);
    atomicAdd(ap + 1, v.y * w);
    atomicAdd(ap + 2, v.z * w);
    atomicAdd(ap + 3, v.w * w);
}

// segment-sum pooling: one wave per node
__global__ void k_pool(const float* __restrict__ h, const int* __restrict__ batch,
                       float* __restrict__ sums, float* __restrict__ cnt, int n) {
    int t = blockIdx.x * blockDim.x + threadIdx.x;
    int node = t >> 5, q = t & 31;
    if (node >= n) return;
    int b = batch[node];
    float4 v = ((const float4*)(h + (size_t)node * 128))[q];
    float* sp = sums + (size_t)b * 128 + (size_t)q * 4;
    atomicAdd(sp + 0, v.x);
    atomicAdd(sp + 1, v.y);
    atomicAdd(sp + 2, v.z);
    atomicAdd(sp + 3, v.w);
    if (q == 0) atomicAdd(&cnt[b], 1.0f);
}

// out[b,o] = (sums[b,:]/max(cnt,1)) . linW[:,o] + linb[o]
__global__ void k_final(const float* __restrict__ sums, const float* __restrict__ cnt,
                        const float* __restrict__ linW, const float* __restrict__ linb,
                        float* __restrict__ out, int B, int O) {
    int t = blockIdx.x * blockDim.x + threadIdx.x;
    if (t >= B * O) return;
    int b = t / O, o = t % O;
    float inv = 1.0f / fmaxf(cnt[b], 1.0f);
    const float* row = sums + (size_t)b * 128;
    float acc = linb[o];
#pragma unroll 4
    for (int k = 0; k < 128; ++k) acc = fmaf(row[k] * inv, linW[k * O + o], acc);
    out[t] = acc;
}

// ---------------- launcher ----------------

extern "C" void kernel_launch(void* const* d_in, const int* in_sizes, int n_in,
                              void* d_out, int out_size, void* d_ws, size_t ws_size,
                              hipStream_t stream) {
    const int*   x     = (const int*)  d_in[0];
    const int*   eidx  = (const int*)  d_in[1];
    // d_in[2] = edge_attr : unused by the reference
    const int*   batch = (const int*)  d_in[3];
    const float* embed = (const float*)d_in[4];
    const float* W1 = (const float*)d_in[5];   const float* b1 = (const float*)d_in[6];
    const float* W2 = (const float*)d_in[7];   const float* b2 = (const float*)d_in[8];
    const float* W3 = (const float*)d_in[9];   const float* b3 = (const float*)d_in[10];
    const float* linW = (const float*)d_in[11]; const float* linb = (const float*)d_in[12];
    float* out = (float*)d_out;

    const int N = in_sizes[0];
    const int E = in_sizes[1] / 2;
    const int O = in_sizes[12];
    const int B = out_size / O;

    const int* src = eidx;        // edge_index[0, :]
    const int* dst = eidx + E;    // edge_index[1, :]

    char* ws = (char*)d_ws;
    const size_t NB = (size_t)N * 128 * sizeof(float);
    float* buf0 = (float*)(ws);                 // node features (ping)
    float* buf1 = (float*)(ws + NB);            // hw scratch
    float* buf2 = (float*)(ws + 2 * NB);        // node features (pong)
    float* dinv = (float*)(ws + 3 * NB);        // N floats (deg -> dinv in place)
    size_t doff = 3 * NB + (((size_t)N * 4 + 255) / 256) * 256;
    float* sums = (float*)(ws + doff);          // B*128 floats
    float* cnt  = sums + (size_t)B * 128;       // B floats (contiguous with sums)

    auto cdiv = [](long a, long b) { return (int)((a + b - 1) / b); };

    // degree / normalization (same for all three layers)
    k_zero<<<cdiv(N, TPB), TPB, 0, stream>>>(dinv, N);
    k_deg <<<cdiv(E, TPB), TPB, 0, stream>>>(dst, dinv, E);
    k_dinv<<<cdiv(N, TPB), TPB, 0, stream>>>(dinv, N);

    // h0 = embed[x]
    k_gather<<<cdiv((long)N * 32, TPB), TPB, 0, stream>>>(x, embed, buf0, N);

    const int gemm_blocks = cdiv(cdiv(N, 16), 8);
    struct Layer { const float* Wm; const float* bm; float* hin; float* aggb; int relu; };
    Layer L[3] = {
        { W1, b1, buf0, buf2, 0 },
        { W2, b2, buf2, buf0, 1 },  // ReLU of previous layer fused into GEMM A-load
        { W3, b3, buf0, buf2, 1 },
    };
    for (int l = 0; l < 3; ++l) {
        if (L[l].relu)
            k_gemm128<1><<<gemm_blocks, TPB, 0, stream>>>(L[l].hin, L[l].Wm, buf1, N);
        else
            k_gemm128<0><<<gemm_blocks, TPB, 0, stream>>>(L[l].hin, L[l].Wm, buf1, N);
        k_agg_init<<<cdiv((long)N * 32, TPB), TPB, 0, stream>>>(buf1, dinv, L[l].bm, L[l].aggb, N);
        k_scatter<<<cdiv((long)E * 32, TPB), TPB, 0, stream>>>(src, dst, dinv, buf1, L[l].aggb, E);
    }

    // mean pooling + linear head (layer-3 output in buf2, no ReLU per reference)
    k_zero<<<cdiv(B * 128 + B, TPB), TPB, 0, stream>>>(sums, B * 128 + B);
    k_pool<<<cdiv((long)N * 32, TPB), TPB, 0, stream>>>(buf2, batch, sums, cnt, N);
    k_final<<<cdiv(B * O, TPB), TPB, 0, stream>>>(sums, cnt, linW, linb, out, B, O);
}